// CAMSAM_89172111000110
// MI455X (gfx1250) — compile-verified
//
#include <hip/hip_runtime.h>
#include <cstdint>

// ---------------- CDNA5 (gfx1250) specifics ----------------
#define AS1 __attribute__((address_space(1)))
#define AS3 __attribute__((address_space(3)))

typedef __attribute__((ext_vector_type(2))) float v2f;
typedef __attribute__((ext_vector_type(8))) float v8f;

// Async-LDS builtin signature (probe-verified round 1):
//   (int4 __device__* gsrc, int4 __shared__* ldst, imm offset, imm cpol)
typedef int v4i __attribute__((vector_size(16)));
typedef AS1 v4i g_v4i;
typedef AS3 v4i l_v4i;

#ifndef __has_builtin
#define __has_builtin(x) 0
#endif

#if __has_builtin(__builtin_amdgcn_global_load_async_to_lds_b128)
#define HAVE_ASYNC_LDS 1
#else
#define HAVE_ASYNC_LDS 0
#endif

static __device__ __forceinline__ void wait_async0() {
#if __has_builtin(__builtin_amdgcn_s_wait_asynccnt)
  __builtin_amdgcn_s_wait_asynccnt(0);
#else
  asm volatile("s_wait_asynccnt 0" ::: "memory");
#endif
}

// ---------------- kernel ----------------
// One workgroup per (n,c) 80x80 plane. 256 threads = 8 wave32.
// LDS: x plane, horizontal box sums of x and x^2, energy accumulator (100 KB).
#define HW 80
#define PLANE (HW * HW) // 6400

__global__ __launch_bounds__(256) void simam_energy_kernel(
    const float* __restrict__ x, float* __restrict__ out, int planes) {
  __shared__ float xs[PLANE];
  __shared__ float hx[PLANE];
  __shared__ float hx2[PLANE];
  __shared__ float Eacc[PLANE];

  const int tid = threadIdx.x;
  // Wave-uniform wave id: forces the WMMA tile loops into scalar (SGPR)
  // control flow so EXEC is provably all-ones at every v_wmma (ISA req).
  const int wave = __builtin_amdgcn_readfirstlane(tid >> 5);
  const int lane = tid & 31;
  const int half = lane >> 4;   // 0: lanes 0-15, 1: lanes 16-31
  const int mrow = lane & 15;

  const float* xg = x + (size_t)blockIdx.x * PLANE;
  float* og = out + (size_t)blockIdx.x * PLANE;

  // ---- stage the plane into LDS (CDNA5 async global->LDS path) ----
#if HAVE_ASYNC_LDS
  for (int i = tid; i < PLANE / 4; i += 256) {
    g_v4i* gp = (g_v4i*)(unsigned long long)(xg + 4 * i);
    l_v4i* lp = (l_v4i*)(unsigned)(unsigned long long)&xs[4 * i];
    __builtin_amdgcn_global_load_async_to_lds_b128(gp, lp, 0, 0);
  }
  wait_async0();
#else
  for (int i = tid; i < PLANE / 4; i += 256) {
    const float4 v = ((const float4*)xg)[i];
    *(float4*)&xs[4 * i] = v;
  }
#endif
  __syncthreads();

#pragma unroll
  for (int ki = 0; ki < 3; ++ki) {
    const int k = (ki == 0) ? 3 : (ki == 1) ? 7 : 11;
    const int p = k >> 1;                      // compile-time constant
    const float invk2 = 1.0f / (float)(k * k);

    // ---- horizontal zero-padded box sums of x and x^2 (scalar, conflict-free)
    for (int idx = tid; idx < PLANE; idx += 256) {
      const int r = idx / HW;
      const int c = idx - r * HW;
      float sx = 0.0f, sx2 = 0.0f;
#pragma unroll
      for (int d = -p; d <= p; ++d) {
        const int cc = c + d;
        if ((unsigned)cc < (unsigned)HW) {
          const float v = xs[r * HW + cc];
          sx += v;
          sx2 = fmaf(v, v, sx2);
        }
      }
      hx[idx] = sx;
      hx2[idx] = sx2;
    }
    __syncthreads();

    // ---- vertical pass as banded-matrix WMMA:  V = Band_k(80x80) * H
    // 5x5 output tiles of 16x16, K blocked by 4 (V_WMMA_F32_16X16X4_F32).
    // A (banded 0/1) is synthesized per-lane; B rows come from LDS.
    // All control flow here is wave-uniform (derived from `wave`).
    for (int t = wave; t < 25; t += 8) {
      const int mi = t / 5;
      const int nj = t - mi * 5;
      const int M0 = mi * 16;
      const int N0 = nj * 16;
      const int N = N0 + mrow; // column owned by this lane for B and D

      int kmin = M0 - p; if (kmin < 0) kmin = 0;
      int kmax = M0 + 15 + p; if (kmax > HW - 1) kmax = HW - 1;
      const int kb0 = kmin >> 2;
      const int kb1 = kmax >> 2;

      v8f cx = {};
      v8f c2 = {};

      for (int kb = kb0; kb <= kb1; ++kb) {
        const int k0 = kb * 4 + half * 2;
        const int k1 = k0 + 1;
        const int Mg = M0 + mrow;
        // banded A element: 1 iff |row - k| <= p and k inside matrix
        const float a0 = (k0 < HW && k0 >= Mg - p && k0 <= Mg + p) ? 1.0f : 0.0f;
        const float a1 = (k1 < HW && k1 >= Mg - p && k1 <= Mg + p) ? 1.0f : 0.0f;
        const int k0c = (k0 < HW) ? k0 : 0;
        const int k1c = (k1 < HW) ? k1 : 0;

        v2f A;  A[0] = a0;                 A[1] = a1;
        v2f Bx; Bx[0] = hx[k0c * HW + N];  Bx[1] = hx[k1c * HW + N];
        v2f B2; B2[0] = hx2[k0c * HW + N]; B2[1] = hx2[k1c * HW + N];

        cx = __builtin_amdgcn_wmma_f32_16x16x4_f32(
            false, A, false, Bx, (short)0, cx, false, false);
        c2 = __builtin_amdgcn_wmma_f32_16x16x4_f32(
            false, A, false, B2, (short)0, c2, false, false);
      }

      // D layout: VGPR i -> row M0 + i + 8*half, col N. Compute energy term.
#pragma unroll
      for (int i = 0; i < 8; ++i) {
        const int Mg = M0 + i + 8 * half;
        const int idx = Mg * HW + N;
        const float mu = cx[i] * invk2;
        const float ex2 = c2[i] * invk2;
        const float var = ex2 - mu * mu;
        const float xv = xs[idx];
        const float d = xv - mu;
        const float term = d * d / (4.0f * (var + 1e-4f)) + 0.5f;
        const float prev = (ki == 0) ? 0.0f : Eacc[idx];
        Eacc[idx] = prev + term;
      }
    }
    __syncthreads();
  }

  // ---- finalize: out = x * sigmoid(1 - E/3)
  for (int idx = tid; idx < PLANE; idx += 256) {
    const float e = Eacc[idx] * (1.0f / 3.0f);
    const float att = 1.0f / (1.0f + __expf(e - 1.0f));
    og[idx] = xs[idx] * att;
  }
}

// ---------------- launch ----------------
extern "C" void kernel_launch(void* const* d_in, const int* in_sizes, int n_in,
                              void* d_out, int out_size, void* d_ws, size_t ws_size,
                              hipStream_t stream) {
  (void)n_in; (void)out_size; (void)d_ws; (void)ws_size;
  const float* x = (const float*)d_in[0];
  float* out = (float*)d_out;
  const int planes = in_sizes[0] / PLANE; // (16*256) = 4096 for the reference shape
  if (planes <= 0) return;
  simam_energy_kernel<<<dim3(planes), dim3(256), 0, stream>>>(x, out, planes);
}